// NeuralStatisticalGate_61641370632840
// MI455X (gfx1250) — compile-verified
//
#include <hip/hip_runtime.h>

#define WINDOW 50

typedef __attribute__((ext_vector_type(2))) float v2f;
typedef __attribute__((ext_vector_type(8))) float v8f;

// One wave (32 threads) handles 16 consecutive windows of one batch row.
// MLP 4->64->32->2 done with V_WMMA_F32_16X16X4_F32 (full fp32 matrix path).
__global__ __launch_bounds__(32)
void nsg_kernel(const float* __restrict__ x,
                const float* __restrict__ w1, const float* __restrict__ b1,
                const float* __restrict__ w2, const float* __restrict__ b2,
                const float* __restrict__ w3, const float* __restrict__ b3,
                float* __restrict__ out, int N, int M)
{
    __shared__ __attribute__((aligned(16))) float xtile[80];        // 65 used
    __shared__ __attribute__((aligned(16))) float statsL[16][4];    // mean,std,skew,kurt
    __shared__ __attribute__((aligned(16))) float h1[16][66];       // 16x64, padded stride
    __shared__ __attribute__((aligned(16))) float h2[16][34];       // 16x32, padded stride
    __shared__ __attribute__((aligned(16))) float paramsL[16][2];   // gamma,beta

    const int lane = threadIdx.x;
    const int b    = blockIdx.y;
    const int m0   = blockIdx.x * 16;
    const float* xb = x + (size_t)b * (size_t)N;

    // ---- load x tile covering windows m0..m0+15 (needs indices m0..m0+64) ----
    for (int i = lane; i < 65; i += 32) {
        int g = m0 + i;
        xtile[i] = (g < N) ? xb[g] : 0.0f;
    }
    __syncthreads();

    // ---- rolling stats: lanes 0..15, one window each (two-pass, matches ref) ----
    if (lane < 16) {
        int m = m0 + lane;
        if (m > M - 1) m = M - 1;          // clamp tail (store is masked later)
        const int r0 = m - m0;
        float s1 = 0.0f;
        #pragma unroll
        for (int i = 0; i < WINDOW; ++i) s1 += xtile[r0 + i];
        const float mean = s1 * (1.0f / (float)WINDOW);
        float c2 = 0.0f, c3 = 0.0f, c4 = 0.0f;
        #pragma unroll
        for (int i = 0; i < WINDOW; ++i) {
            float d  = xtile[r0 + i] - mean;
            float d2 = d * d;
            c2 += d2; c3 += d2 * d; c4 += d2 * d2;
        }
        const float var  = c2 * (1.0f / (float)(WINDOW - 1));
        const float stdv = sqrtf(var) + 1e-6f;
        const float inv3 = 1.0f / (stdv * stdv * stdv);
        statsL[lane][0] = mean;
        statsL[lane][1] = stdv;
        statsL[lane][2] = c3 * (1.0f / (float)WINDOW) * inv3;          // skew
        statsL[lane][3] = c4 * (1.0f / (float)WINDOW) * (inv3 / stdv); // kurt
    }
    __syncthreads();

    // Per-lane WMMA layout helpers (wave32):
    //  A 16x4 f32: lane -> row M=lane%16, VGPR0/1 -> K = (lane/16)*2 + {0,1}
    //  B  4x16 f32: lane -> col N=lane%16, VGPR0/1 -> K = (lane/16)*2 + {0,1}
    //  C/D 16x16 f32 (v8f): VGPR j -> M = j + (lane/16)*8, N = lane%16
    const int colL    = lane & 15;
    const int khalf   = (lane >> 4) << 1;   // 0 or 2
    const int rowBase = (lane >> 4) << 3;   // 0 or 8

    // ================= layer 1: stats(16x4) @ w1(4x64) + b1, ReLU =================
    v2f a1;
    a1.x = statsL[colL][khalf];
    a1.y = statsL[colL][khalf + 1];
    #pragma unroll
    for (int n = 0; n < 4; ++n) {
        v2f bt;
        bt.x = w1[(khalf    ) * 64 + 16 * n + colL];
        bt.y = w1[(khalf + 1) * 64 + 16 * n + colL];
        v8f acc = {};
        acc = __builtin_amdgcn_wmma_f32_16x16x4_f32(false, a1, false, bt,
                                                    (short)0, acc, false, false);
        const float bias = b1[16 * n + colL];
        #pragma unroll
        for (int j = 0; j < 8; ++j) {
            float v = acc[j] + bias;
            h1[rowBase + j][16 * n + colL] = v > 0.0f ? v : 0.0f;
        }
    }
    __syncthreads();

    // ================= layer 2: h1(16x64) @ w2(64x32) + b2, ReLU =================
    v8f acc2[2];
    acc2[0] = (v8f){};
    acc2[1] = (v8f){};
    #pragma unroll
    for (int kc = 0; kc < 16; ++kc) {
        v2f a;
        a.x = h1[colL][4 * kc + khalf];
        a.y = h1[colL][4 * kc + khalf + 1];
        #pragma unroll
        for (int n = 0; n < 2; ++n) {
            v2f bt;
            bt.x = w2[(4 * kc + khalf    ) * 32 + 16 * n + colL];
            bt.y = w2[(4 * kc + khalf + 1) * 32 + 16 * n + colL];
            acc2[n] = __builtin_amdgcn_wmma_f32_16x16x4_f32(false, a, false, bt,
                                                            (short)0, acc2[n], false, false);
        }
    }
    #pragma unroll
    for (int n = 0; n < 2; ++n) {
        const float bias = b2[16 * n + colL];
        #pragma unroll
        for (int j = 0; j < 8; ++j) {
            float v = acc2[n][j] + bias;
            h2[rowBase + j][16 * n + colL] = v > 0.0f ? v : 0.0f;
        }
    }
    __syncthreads();

    // ================= layer 3: h2(16x32) @ w3(32x2) + b3 (N padded to 16) ========
    v8f acc3 = {};
    #pragma unroll
    for (int kc = 0; kc < 8; ++kc) {
        v2f a;
        a.x = h2[colL][4 * kc + khalf];
        a.y = h2[colL][4 * kc + khalf + 1];
        v2f bt;
        if (colL < 2) {
            bt.x = w3[(4 * kc + khalf    ) * 2 + colL];
            bt.y = w3[(4 * kc + khalf + 1) * 2 + colL];
        } else {
            bt.x = 0.0f;
            bt.y = 0.0f;
        }
        acc3 = __builtin_amdgcn_wmma_f32_16x16x4_f32(false, a, false, bt,
                                                     (short)0, acc3, false, false);
    }
    if (colL < 2) {
        const float bias = b3[colL];
        #pragma unroll
        for (int j = 0; j < 8; ++j)
            paramsL[rowBase + j][colL] = acc3[j] + bias;
    }
    __syncthreads();

    // ================= reproject center element and store =================
    if (lane < 16) {
        const int m = m0 + lane;
        if (m < M) {
            const float mean  = statsL[lane][0];
            const float stdv  = statsL[lane][1];
            const float gamma = paramsL[lane][0];
            const float beta  = paramsL[lane][1];
            const float xc    = xtile[lane + WINDOW / 2];
            out[(size_t)b * (size_t)M + m] = (xc - mean) / stdv * gamma + beta;
        }
    }
}

extern "C" void kernel_launch(void* const* d_in, const int* in_sizes, int n_in,
                              void* d_out, int out_size, void* d_ws, size_t ws_size,
                              hipStream_t stream) {
    const float* x  = (const float*)d_in[0];
    const float* w1 = (const float*)d_in[1];
    const float* b1 = (const float*)d_in[2];
    const float* w2 = (const float*)d_in[3];
    const float* b2 = (const float*)d_in[4];
    const float* w3 = (const float*)d_in[5];
    const float* b3 = (const float*)d_in[6];
    float* out = (float*)d_out;

    // in_sizes[0] = B*N, out_size = B*(N-49)  =>  B = (B*N - B*M)/49
    int B = (in_sizes[0] - out_size) / (WINDOW - 1);
    if (B <= 0) B = 32;
    int N = in_sizes[0] / B;
    int M = N - WINDOW + 1;

    dim3 grid((M + 15) / 16, B);
    nsg_kernel<<<grid, 32, 0, stream>>>(x, w1, b1, w2, b2, w3, b3, out, N, M);
}